// Encoder_65077344469353
// MI455X (gfx1250) — compile-verified
//
#include <hip/hip_runtime.h>

// Positional encoder: out[n, c*20 + {0:sin,1:cos}*10 + d] = sin/cos(x[n,c] * 2^d)
// N = 2e6 rows, 3 channels, D_MAX = 10  ->  60 f32 outputs per row.
//
// Memory-bound: 24 MB load + 480 MB store ~= 22 us @ 23.3 TB/s.
//   * HW v_sin/v_cos only at base frequency (small args -> accurate),
//     then 9 double-angle steps per channel (s'=2sc, c'=1-2s^2).
//   * 256x60 f32 output tile staged in LDS (layout == output layout),
//     drained with coalesced b128 non-temporal stores (write-once stream,
//     480 MB >> 192 MB L2 -> TH=NT is the right policy).
//   * Inputs staged via gfx1250 async global->LDS loads (ASYNCcnt path).

#define DMAX   10
#define ROWS   256           // rows per block (= blockDim.x)
#define RLEN   60            // f32 per output row (3 ch * 2 * DMAX)

typedef __attribute__((ext_vector_type(4))) float v4f;   // native vec -> b128

#if defined(__gfx1250__) && \
    __has_builtin(__builtin_amdgcn_global_load_async_to_lds_b32) && \
    __has_builtin(__builtin_amdgcn_s_wait_asynccnt)
#define PE_USE_ASYNC_LDS 1
typedef __attribute__((address_space(1))) int* pe_gptr;
typedef __attribute__((address_space(3))) int* pe_lptr;
#endif

__device__ __forceinline__ void nt_store4(v4f* p, v4f v) {
  __builtin_nontemporal_store(v, p);    // global_store_b128 ... th:TH_STORE_NT
}

__global__ __launch_bounds__(ROWS) void Encoder_65077344469353_kernel(
    const float* __restrict__ x, float* __restrict__ out, int nrows) {
  __shared__ __align__(16) float sIn[ROWS * 3];
  __shared__ __align__(16) float sOut[ROWS * RLEN];

  const int t    = threadIdx.x;
  const int r0   = blockIdx.x * ROWS;
  const int rows = min(ROWS, nrows - r0);        // tail block: rows < 256

  // ---------------- stage inputs (3*rows dwords) into LDS ----------------
#if PE_USE_ASYNC_LDS
  for (int i = t; i < rows * 3; i += ROWS) {
    __builtin_amdgcn_global_load_async_to_lds_b32(
        (pe_gptr)(x + (size_t)r0 * 3 + i), (pe_lptr)(&sIn[i]),
        /*offset=*/0, /*cpol=*/0);
  }
  __builtin_amdgcn_s_wait_asynccnt(0);   // wave's own async copies done
  __syncthreads();                        // other waves' copies visible
#else
  for (int i = t; i < rows * 3; i += ROWS)
    sIn[i] = __builtin_nontemporal_load(x + (size_t)r0 * 3 + i);
  __syncthreads();
#endif

  // ---------------- compute one row per thread into LDS tile -------------
  if (t < rows) {
    float* dst = &sOut[t * RLEN];
#pragma unroll
    for (int c = 0; c < 3; ++c) {
      const float a = sIn[3 * t + c];
      // base frequency via HW trans ops (args are ~N(0,1): accurate range)
      float s  = __sinf(a);
      float co = __cosf(a);
      float buf[2 * DMAX];
#pragma unroll
      for (int d = 0; d < DMAX; ++d) {
        buf[d]        = s;    // sin(2^d * a)
        buf[DMAX + d] = co;   // cos(2^d * a)
        const float s2 = 2.0f * s * co;           // sin(2a) = 2 sin cos
        const float c2 = fmaf(-2.0f * s, s, 1.0f);// cos(2a) = 1 - 2 sin^2
        s  = s2;
        co = c2;
      }
      // 20 contiguous floats -> 5 x b128 LDS stores (16B aligned: 240|16, 80|16)
      v4f* q = (v4f*)(dst + c * 2 * DMAX);
#pragma unroll
      for (int i = 0; i < 5; ++i) {
        v4f v = {buf[4 * i + 0], buf[4 * i + 1], buf[4 * i + 2], buf[4 * i + 3]};
        q[i] = v;
      }
    }
  }
  __syncthreads();

  // ---------------- drain tile: LDS layout == output layout --------------
  // rows*60 floats = rows*15 float4; fully coalesced b128 NT stores.
  const v4f* s4 = (const v4f*)sOut;
  v4f*       o4 = (v4f*)(out + (size_t)r0 * RLEN);
  if (rows == ROWS) {
    // full block: compile-time trip count -> all ds_load_b128 hoisted,
    // store stream issues back-to-back
#pragma unroll
    for (int i = 0; i < RLEN / 4; ++i)   // 15 iters
      nt_store4(&o4[t + i * ROWS], s4[t + i * ROWS]);
  } else {
    const int total4 = rows * (RLEN / 4);
    for (int i = t; i < total4; i += ROWS) nt_store4(&o4[i], s4[i]);
  }
}

extern "C" void kernel_launch(void* const* d_in, const int* in_sizes, int n_in,
                              void* d_out, int out_size, void* d_ws, size_t ws_size,
                              hipStream_t stream) {
  const float* x   = (const float*)d_in[0];
  float*       out = (float*)d_out;
  const int nrows  = in_sizes[0] / 3;          // 2,000,000
  const int blocks = (nrows + ROWS - 1) / ROWS;
  Encoder_65077344469353_kernel<<<blocks, ROWS, 0, stream>>>(x, out, nrows);
}